// TemporalComplementarityScorer_22239340659064
// MI455X (gfx1250) — compile-verified
//
#include <hip/hip_runtime.h>

// ---------------------------------------------------------------------------
// TemporalComplementarityScorer for MI455X (gfx1250, wave32).
//
// Stage GEMMs (192->64->32->2x32) use V_WMMA_F32_16X16X4_F32 in full fp32.
// Pair stage (dominant, ReLU-nonlinear -> not a GEMM) is a tiled VALU kernel
// that computes each symmetric pair once and writes both halves coalesced.
// ---------------------------------------------------------------------------

typedef float v2f __attribute__((ext_vector_type(2)));
typedef float v8f __attribute__((ext_vector_type(8)));

#define EMBED 128
#define TEMPO 64
#define KDIM  192      // EMBED + TEMPO
#define H1D   64
#define H2D   32
#define NB    4
#define NN    1024
#define MROWS (NB * NN)   // 4096

// D = A(16x4 f32) * B(4x16 f32) + C(16x16 f32)
__device__ __forceinline__ v8f wmma4(v2f a, v2f b, v8f c) {
  return __builtin_amdgcn_wmma_f32_16x16x4_f32(
      /*neg_a=*/false, a, /*neg_b=*/false, b,
      /*c_mod=*/(short)0, c, /*reuse_a=*/false, /*reuse_b=*/false);
}

// A-fragment 16x4 (MxK), row-major source with leading dim ld.
// ISA layout: lane l holds row m=l&15; lanes 0-15 -> K={k0,k0+1} in vgpr{0,1},
// lanes 16-31 -> K={k0+2,k0+3}.
__device__ __forceinline__ v2f load_a_frag(const float* src, int ld, int k0, int lane) {
  int m = lane & 15;
  int kk = k0 + ((lane >> 4) << 1);
  const float* p = src + (size_t)m * ld + kk;
  v2f r = { p[0], p[1] };
  return r;
}

// A-fragment from the virtual concat [embeddings | temporal] (ld 128 / 64).
__device__ __forceinline__ v2f load_x_frag(const float* emb, const float* tmp,
                                           int row0, int k0, int lane) {
  int m = lane & 15;
  int kk = k0 + ((lane >> 4) << 1);       // kk even; chunk never straddles 128
  const float* p = (kk < EMBED) ? (emb + (size_t)(row0 + m) * EMBED + kk)
                                : (tmp + (size_t)(row0 + m) * TEMPO + (kk - EMBED));
  v2f r = { p[0], p[1] };
  return r;
}

// B-fragment 4x16 (KxN), row-major W with leading dim ld.
// Mirror of A: lane l holds col n=l&15; lanes 0-15 -> K={k0,k0+1},
// lanes 16-31 -> K={k0+2,k0+3}.
__device__ __forceinline__ v2f load_b_frag(const float* W, int ld, int k0, int n0, int lane) {
  int n = n0 + (lane & 15);
  int kk = k0 + ((lane >> 4) << 1);
  v2f r = { W[(size_t)kk * ld + n], W[(size_t)(kk + 1) * ld + n] };
  return r;
}

// ---------------------------------------------------------------------------
// Kernel 1: fused stages 1-3.  One wave = 16 rows; 8 waves/block -> 128 rows.
// Emits:  a_ws = patterns @ W3[:32]        (hi)
//         c_ws = patterns @ W3[32:] + b3   (hj + b3 folded)
// ---------------------------------------------------------------------------
__global__ __launch_bounds__(256)
void features_kernel(const float* __restrict__ emb, const float* __restrict__ tmp,
                     const float* __restrict__ W1, const float* __restrict__ b1,
                     const float* __restrict__ W2, const float* __restrict__ b2,
                     const float* __restrict__ W3, const float* __restrict__ b3,
                     float* __restrict__ a_ws, float* __restrict__ c_ws) {
  __shared__ float sH[8][16 * H1D];   // per-wave 16x64 staging (32 KB)
  __shared__ float sP[8][16 * H2D];   // per-wave 16x32 staging (16 KB)

  const int lane = threadIdx.x & 31;
  const int wave = threadIdx.x >> 5;
  const int row0 = blockIdx.x * 128 + wave * 16;
  const int n = lane & 15;
  const int h = lane >> 4;

  const v8f vzero = { 0.f, 0.f, 0.f, 0.f, 0.f, 0.f, 0.f, 0.f };

  // ---- Stage 1: H = relu(X @ W1 + b1), X = [emb|tmp] (16 x 192 per wave) ----
  v8f acc1[4];
  #pragma unroll
  for (int t = 0; t < 4; ++t) acc1[t] = vzero;
  for (int k0 = 0; k0 < KDIM; k0 += 4) {
    v2f a = load_x_frag(emb, tmp, row0, k0, lane);
    #pragma unroll
    for (int t = 0; t < 4; ++t) {
      v2f b = load_b_frag(W1, H1D, k0, 16 * t, lane);
      acc1[t] = wmma4(a, b, acc1[t]);
    }
  }
  // D layout: lane l, vgpr v -> row (v + 8*(l>>4)), col (l&15) within tile.
  #pragma unroll
  for (int t = 0; t < 4; ++t) {
    float bias = b1[16 * t + n];
    #pragma unroll
    for (int v = 0; v < 8; ++v)
      sH[wave][(v + 8 * h) * H1D + 16 * t + n] = fmaxf(acc1[t][v] + bias, 0.f);
  }
  __syncthreads();

  // ---- Stage 2: P = relu(H @ W2 + b2) (16 x 32) ----
  v8f acc2[2];
  acc2[0] = vzero; acc2[1] = vzero;
  for (int k0 = 0; k0 < H1D; k0 += 4) {
    v2f a = load_a_frag(&sH[wave][0], H1D, k0, lane);
    #pragma unroll
    for (int t = 0; t < 2; ++t) {
      v2f b = load_b_frag(W2, H2D, k0, 16 * t, lane);
      acc2[t] = wmma4(a, b, acc2[t]);
    }
  }
  #pragma unroll
  for (int t = 0; t < 2; ++t) {
    float bias = b2[16 * t + n];
    #pragma unroll
    for (int v = 0; v < 8; ++v)
      sP[wave][(v + 8 * h) * H2D + 16 * t + n] = fmaxf(acc2[t][v] + bias, 0.f);
  }
  __syncthreads();

  // ---- Stage 3: a = P @ W3[:32],  c = P @ W3[32:] + b3 ----
  v8f accA[2], accC[2];
  accA[0] = vzero; accA[1] = vzero; accC[0] = vzero; accC[1] = vzero;
  for (int k0 = 0; k0 < H2D; k0 += 4) {
    v2f a = load_a_frag(&sP[wave][0], H2D, k0, lane);
    #pragma unroll
    for (int t = 0; t < 2; ++t) {
      v2f bA = load_b_frag(W3,               H2D, k0, 16 * t, lane);  // rows 0..31
      v2f bC = load_b_frag(W3 + 32 * H2D,    H2D, k0, 16 * t, lane);  // rows 32..63
      accA[t] = wmma4(a, bA, accA[t]);
      accC[t] = wmma4(a, bC, accC[t]);
    }
  }
  #pragma unroll
  for (int t = 0; t < 2; ++t) {
    float biasC = b3[16 * t + n];
    #pragma unroll
    for (int v = 0; v < 8; ++v) {
      int row = row0 + v + 8 * h;
      a_ws[(size_t)row * H2D + 16 * t + n] = accA[t][v];
      c_ws[(size_t)row * H2D + 16 * t + n] = accC[t][v] + biasC;
    }
  }
}

// ---------------------------------------------------------------------------
// Kernel 2: pairwise stage. One block per upper-triangular 32x32 tile pair.
// out[i][j] = out[j][i] = 0.5*(tanh(f(i,j)) + tanh(f(j,i))),
// f(i,j) = sum_k relu(a[i][k] + c[j][k]) * W4[k] + b4.
// ---------------------------------------------------------------------------
__global__ __launch_bounds__(256)
void pair_kernel(const float* __restrict__ A, const float* __restrict__ C,
                 const float* __restrict__ W4, const float* __restrict__ B4,
                 float* __restrict__ out) {
  const int T = NN / 32;   // 32 tiles per dim
  __shared__ float sAI[32 * 32], sCI[32 * 32];
  __shared__ float sAJ[32 * 32], sCJ[32 * 32];
  __shared__ float sOut[32 * 32];

  // Decode upper-triangular tile pair (ti <= tj) from linear blockIdx.x.
  int ti = 0, rem = blockIdx.x, span = T;
  while (rem >= span) { rem -= span; ++ti; --span; }
  int tj = ti + rem;

  const int b  = blockIdx.y;
  const int i0 = ti * 32, j0 = tj * 32;
  const size_t base = (size_t)b * NN * H2D;

  const int r  = threadIdx.x >> 3;   // 0..31 row within tile
  const int ch = threadIdx.x & 7;    // 0..7 float4 chunk

  // Cooperative tile loads (each row = 32 floats = 8 float4).
  ((float4*)sAI)[r * 8 + ch] = ((const float4*)(A + base + (size_t)(i0 + r) * H2D))[ch];
  ((float4*)sCI)[r * 8 + ch] = ((const float4*)(C + base + (size_t)(i0 + r) * H2D))[ch];
  ((float4*)sAJ)[r * 8 + ch] = ((const float4*)(A + base + (size_t)(j0 + r) * H2D))[ch];
  ((float4*)sCJ)[r * 8 + ch] = ((const float4*)(C + base + (size_t)(j0 + r) * H2D))[ch];
  const float b4v = B4[0];
  __syncthreads();

  const int il = threadIdx.x >> 3;   // thread owns one i (0..31)
  const int jg = threadIdx.x & 7;    // and 4 j's: jg, jg+8, jg+16, jg+24

  float4 ai[8], ci[8], w[8];
  #pragma unroll
  for (int q = 0; q < 8; ++q) {
    ai[q] = ((const float4*)sAI)[il * 8 + q];
    ci[q] = ((const float4*)sCI)[il * 8 + q];
    w[q]  = ((const float4*)W4)[q];          // 128 B, L0-resident
  }

  #pragma unroll
  for (int jj = 0; jj < 4; ++jj) {
    const int jl = jg + 8 * jj;
    float sij = 0.f, sji = 0.f;
    #pragma unroll
    for (int q = 0; q < 8; ++q) {
      float4 cj = ((const float4*)sCJ)[jl * 8 + q];
      float4 aj = ((const float4*)sAJ)[jl * 8 + q];
      sij += fmaxf(ai[q].x + cj.x, 0.f) * w[q].x;
      sji += fmaxf(aj.x + ci[q].x, 0.f) * w[q].x;
      sij += fmaxf(ai[q].y + cj.y, 0.f) * w[q].y;
      sji += fmaxf(aj.y + ci[q].y, 0.f) * w[q].y;
      sij += fmaxf(ai[q].z + cj.z, 0.f) * w[q].z;
      sji += fmaxf(aj.z + ci[q].z, 0.f) * w[q].z;
      sij += fmaxf(ai[q].w + cj.w, 0.f) * w[q].w;
      sji += fmaxf(aj.w + ci[q].w, 0.f) * w[q].w;
    }
    sOut[il * 32 + jl] = 0.5f * (tanhf(sij + b4v) + tanhf(sji + b4v));
  }
  __syncthreads();

  // Coalesced writes of the tile and its transpose.
  float* ob = out + (size_t)b * NN * NN;
  ((float4*)(ob + (size_t)(i0 + r) * NN + j0))[ch] = ((const float4*)sOut)[r * 8 + ch];
  if (ti != tj) {
    float4 tv;
    tv.x = sOut[(4 * ch + 0) * 32 + r];
    tv.y = sOut[(4 * ch + 1) * 32 + r];
    tv.z = sOut[(4 * ch + 2) * 32 + r];
    tv.w = sOut[(4 * ch + 3) * 32 + r];
    ((float4*)(ob + (size_t)(j0 + r) * NN + i0))[ch] = tv;
  }
}

// ---------------------------------------------------------------------------
extern "C" void kernel_launch(void* const* d_in, const int* in_sizes, int n_in,
                              void* d_out, int out_size, void* d_ws, size_t ws_size,
                              hipStream_t stream) {
  const float* emb = (const float*)d_in[0];
  const float* tmp = (const float*)d_in[1];
  const float* W1  = (const float*)d_in[2];
  const float* b1  = (const float*)d_in[3];
  const float* W2  = (const float*)d_in[4];
  const float* b2  = (const float*)d_in[5];
  const float* W3  = (const float*)d_in[6];
  const float* b3  = (const float*)d_in[7];
  const float* W4  = (const float*)d_in[8];
  const float* b4  = (const float*)d_in[9];
  float* out = (float*)d_out;

  float* a_ws = (float*)d_ws;                    // [4096][32] hi
  float* c_ws = a_ws + (size_t)MROWS * H2D;      // [4096][32] hj + b3

  features_kernel<<<MROWS / 128, 256, 0, stream>>>(emb, tmp, W1, b1, W2, b2, W3, b3,
                                                   a_ws, c_ws);

  const int T = NN / 32;
  dim3 grid(T * (T + 1) / 2, NB);                // 528 upper-tri tile pairs x 4 batches
  pair_kernel<<<grid, 256, 0, stream>>>(a_ws, c_ws, W4, b4, out);
}